// ClassficationAndCRLAndMDCA_326417515025
// MI455X (gfx1250) — compile-verified
//
#include <hip/hip_runtime.h>
#include <stdint.h>
#include <math.h>

#define BB   4096
#define CC   32000
#define NWG  128                 // workgroups in row kernel
#define RPW  (BB / NWG)          // 32 rows per workgroup
#define T1   1024                // threads in row kernel (32 waves)
#define NE   32                  // classes per thread (31 full strides + tail)
#define ROW_BYTES (CC * 4)       // 128000

typedef __attribute__((ext_vector_type(2))) float v2f;
typedef __attribute__((ext_vector_type(8))) float v8f;

// ---- CDNA5 async global->LDS copy (ASYNCcnt path), per CDNA5 ISA ----
__device__ __forceinline__ void async_b128(uint32_t lds_off, unsigned long long gaddr) {
    asm volatile("global_load_async_to_lds_b128 %0, %1, off"
                 :: "v"(lds_off), "v"(gaddr) : "memory");
}
__device__ __forceinline__ void wait_async0() {
    asm volatile("s_wait_asynccnt 0x0" ::: "memory");
}

// =====================  kernel 0: class histogram  =====================
__global__ void zero_counts_kernel(unsigned* counts) {
    int i = blockIdx.x * blockDim.x + threadIdx.x;
    if (i < CC) counts[i] = 0u;
}
__global__ void hist_kernel(const int* __restrict__ targets, unsigned* counts) {
    int i = blockIdx.x * blockDim.x + threadIdx.x;
    if (i < BB) atomicAdd(&counts[targets[i]], 1u);
}

// ==========  kernel 1: stream logits once; softmax stats + column partials  ==========
__global__ __launch_bounds__(T1) void rows_kernel(const float* __restrict__ logits,
                                                  const int* __restrict__ targets,
                                                  float* __restrict__ partial,
                                                  float* __restrict__ conf,
                                                  float* __restrict__ ce) {
    extern __shared__ float smem[];
    float* rowbuf = smem;          // CC floats (128 KB) staging buffer
    float* red    = smem + CC;     // reduction scratch: [0..31] wave partials, [32] max, [33] sum
    const int tid  = threadIdx.x;
    const int lane = tid & 31;
    const int wave = tid >> 5;
    const int wg   = blockIdx.x;
    const uint32_t lds_base = (uint32_t)(uintptr_t)rowbuf;   // low 32 bits of flat addr == LDS offset
    const uint32_t my_off   = (uint32_t)tid * 16u;

    float acc[NE];
#pragma unroll
    for (int j = 0; j < NE; ++j) acc[j] = 0.f;

    const int r0 = wg * RPW;

    // prefetch first row into LDS via async DMA
    {
        unsigned long long g = (unsigned long long)(uintptr_t)(logits + (size_t)r0 * CC);
#pragma unroll
        for (int sg = 0; sg < 8; ++sg) {
            uint32_t off = (uint32_t)sg * 16384u + my_off;
            if (off < (uint32_t)ROW_BYTES) async_b128(lds_base + off, g + off);
        }
    }

    for (int ri = 0; ri < RPW; ++ri) {
        const int r = r0 + ri;
        wait_async0();
        __syncthreads();                       // row r resident in LDS for all waves

        // pull this thread's strided classes into registers
        float v[NE];
#pragma unroll
        for (int j = 0; j < 31; ++j) v[j] = rowbuf[tid + (j << 10)];
        v[31] = (tid < (CC - 31 * 1024)) ? rowbuf[tid + 31744] : -__builtin_inff();

        // target logit (needed before buffer is recycled)
        float xt = 0.f;
        if (tid == 0) xt = rowbuf[targets[r]];
        __syncthreads();                       // everyone done reading rowbuf

        // overlap: DMA next row while we compute on registers
        if (ri + 1 < RPW) {
            unsigned long long g = (unsigned long long)(uintptr_t)(logits + (size_t)(r + 1) * CC);
#pragma unroll
            for (int sg = 0; sg < 8; ++sg) {
                uint32_t off = (uint32_t)sg * 16384u + my_off;
                if (off < (uint32_t)ROW_BYTES) async_b128(lds_base + off, g + off);
            }
        }

        // ---- row max ----
        float lmax = v[0];
#pragma unroll
        for (int j = 1; j < NE; ++j) lmax = fmaxf(lmax, v[j]);
        for (int o = 16; o; o >>= 1) lmax = fmaxf(lmax, __shfl_xor(lmax, o, 32));
        if (lane == 0) red[wave] = lmax;
        __syncthreads();
        if (wave == 0) {
            float x = red[lane];
            for (int o = 16; o; o >>= 1) x = fmaxf(x, __shfl_xor(x, o, 32));
            if (lane == 0) red[32] = x;
        }
        __syncthreads();
        const float m = red[32];

        // ---- row sum of exp ----
        float lsum = 0.f;
#pragma unroll
        for (int j = 0; j < NE; ++j) lsum += __expf(v[j] - m);
        for (int o = 16; o; o >>= 1) lsum += __shfl_xor(lsum, o, 32);
        if (lane == 0) red[wave] = lsum;
        __syncthreads();
        if (wave == 0) {
            float x = red[lane];
            for (int o = 16; o; o >>= 1) x += __shfl_xor(x, o, 32);
            if (lane == 0) red[33] = x;
        }
        __syncthreads();
        const float s = red[33];
        const float inv_s = 1.0f / s;

        if (tid == 0) {
            conf[r] = inv_s;                          // max softmax prob = exp(m-m)/s
            ce[r]   = xt - m - __logf(s);             // log p[target]
        }

        // ---- accumulate column partials of p ----
#pragma unroll
        for (int j = 0; j < NE; ++j) acc[j] += __expf(v[j] - m) * inv_s;
    }

    // write this WG's column partial vector
    float* pw = partial + (size_t)wg * CC;
#pragma unroll
    for (int j = 0; j < 31; ++j) pw[tid + (j << 10)] = acc[j];
    if (tid < (CC - 31 * 1024)) pw[tid + 31744] = acc[31];
}

// ==========  kernel 2: WMMA reduction of partials -> |avg_conf - avg_count|  ==========
__global__ __launch_bounds__(256) void calib_kernel(const float* __restrict__ partial,
                                                    const unsigned* __restrict__ counts,
                                                    float* __restrict__ absdiff) {
    const int tid  = threadIdx.x;
    const int lane = tid & 31;
    const int wv   = (blockIdx.x * blockDim.x + tid) >> 5;   // wave id; uniform per wave
    if (wv >= CC / 16) return;
    const int c0 = wv * 16;
    const int n  = lane & 15;
    const int kb = (lane < 16) ? 0 : 2;
    const float* p0 = partial + (size_t)kb * CC + (size_t)(c0 + n);

    v2f a; a.x = 1.0f; a.y = 1.0f;      // A = ones(16x4): D[m,n] = sum_k B[k,n]
    v8f acc = {};
    for (int w = 0; w < NWG; w += 4) {
        v2f b;
        b.x = p0[(size_t)w * CC];
        b.y = p0[(size_t)(w + 1) * CC];
#if __has_builtin(__builtin_amdgcn_wmma_f32_16x16x4_f32)
        acc = __builtin_amdgcn_wmma_f32_16x16x4_f32(false, a, false, b,
                                                    (short)0, acc, false, false);
#else
        acc[0] += b.x + b.y + __shfl_xor(b.x + b.y, 16, 32);  // fallback (should not trigger)
#endif
    }
    if (lane < 16) {
        const float total = acc[0];      // VGPR0, lanes 0-15 => M=0, N=lane
        const float cnt   = (float)counts[c0 + lane];
        absdiff[c0 + lane] = fabsf((total - cnt) * (1.0f / BB));
    }
}

// ==========  kernel 3: fold everything into the scalar loss  ==========
__global__ __launch_bounds__(1024) void final_kernel(const float* __restrict__ absdiff,
                                                     const float* __restrict__ ce,
                                                     const float* __restrict__ conf,
                                                     const int* __restrict__ idx,
                                                     const float* __restrict__ corr,
                                                     float* __restrict__ out) {
    __shared__ float red[32];
    const int tid = threadIdx.x, lane = tid & 31, wave = tid >> 5;
    float acc = 0.f;
    for (int c = tid; c < CC; c += 1024) acc += absdiff[c] * (1.0f / CC);       // MDCA
    for (int b = tid; b < BB; b += 1024) acc += (-ce[b]) * (1.0f / BB);         // cross entropy
    for (int b = tid; b < BB; b += 1024) {                                      // CRL
        const int   b2 = (b + 1) & (BB - 1);
        const float c1 = corr[idx[b]];
        const float c2 = corr[idx[b2]];
        const float d  = c1 - c2;
        const float t  = (d > 0.f) ? 1.f : ((d < 0.f) ? -1.f : 0.f);
        const float tnz = (t == 0.f) ? 1.f : t;
        const float x2  = conf[b2] + fabsf(d) / tnz;
        acc += fmaxf(0.f, -t * (conf[b] - x2)) * (1.0f / BB);
    }
    for (int o = 16; o; o >>= 1) acc += __shfl_xor(acc, o, 32);
    if (lane == 0) red[wave] = acc;
    __syncthreads();
    if (wave == 0) {
        float x = red[lane];
        for (int o = 16; o; o >>= 1) x += __shfl_xor(x, o, 32);
        if (lane == 0) out[0] = x;
    }
}

extern "C" void kernel_launch(void* const* d_in, const int* in_sizes, int n_in,
                              void* d_out, int out_size, void* d_ws, size_t ws_size,
                              hipStream_t stream) {
    const float* logits  = (const float*)d_in[0];
    const int*   targets = (const int*)d_in[1];
    const int*   idx     = (const int*)d_in[2];
    const float* corr    = (const float*)d_in[3];

    char* ws = (char*)d_ws;
    float*    partial = (float*)ws;                                  // NWG*CC floats
    float*    conf    = (float*)(ws + (size_t)NWG * CC * 4);         // BB floats
    float*    ce      = conf + BB;                                   // BB floats
    float*    absd    = ce + BB;                                     // CC floats
    unsigned* counts  = (unsigned*)(absd + CC);                      // CC uints

    (void)in_sizes; (void)n_in; (void)out_size; (void)ws_size;

    zero_counts_kernel<<<(CC + 255) / 256, 256, 0, stream>>>(counts);
    hist_kernel<<<(BB + 255) / 256, 256, 0, stream>>>(targets, counts);

    const size_t smem = (size_t)CC * 4 + 34 * 4;                     // row buffer + reduce scratch
    static bool attr_set = false;
    if (!attr_set) {   // host-side attribute, not a stream op; deterministic no-op after first call
        (void)hipFuncSetAttribute((const void*)rows_kernel,
                                  hipFuncAttributeMaxDynamicSharedMemorySize, (int)smem);
        attr_set = true;
    }
    rows_kernel<<<NWG, T1, smem, stream>>>(logits, targets, partial, conf, ce);

    calib_kernel<<<(CC / 16) * 32 / 256, 256, 0, stream>>>(partial, counts, absd);
    final_kernel<<<1, 1024, 0, stream>>>(absd, ce, conf, idx, corr, (float*)d_out);
}